// MinimalFirstSpikeWTA_17059610100017
// MI455X (gfx1250) — compile-verified
//
#include <hip/hip_runtime.h>
#include <hip/hip_bf16.h>
#include <stdint.h>

// Problem constants (reference is fixed-shape).
#define B_    64
#define L_    2048
#define K_    512
#define S_    16                    // segments of L per batch (parallelism: 16*64 = 1024 blocks)
#define TROWS 16                    // timestep rows per LDS tile
#define TILES (L_ / (S_ * TROWS))   // 8 tiles per block
#define TILE_BYTES (TROWS * K_ * 4) // 32768 B per tile
#define THRv  0.5f

typedef __attribute__((ext_vector_type(2))) float v2f;
typedef __attribute__((ext_vector_type(4))) float v4f;
typedef __attribute__((ext_vector_type(8))) float v8f;

// Issue one 32KB tile (16 rows x 512 f32) as 4 async b128 loads per thread.
// The
// immediate offset advances BOTH sides, so one (lds,global) pair suffices.
__device__ __forceinline__ void async_tile_load(uint32_t lds, uint64_t g) {
  asm volatile(
      "global_load_async_to_lds_b128 %0, %1, off\n\t"
      "global_load_async_to_lds_b128 %0, %1, off offset:8192\n\t"
      "global_load_async_to_lds_b128 %0, %1, off offset:16384\n\t"
      "global_load_async_to_lds_b128 %0, %1, off offset:24576"
      :: "v"(lds), "v"(g) : "memory");
}

// ---------------- Pass 1: per-(b,k) first-spike time + WMMA column sums ----
__global__ __launch_bounds__(512) void fs_pass1(const float* __restrict__ spikes,
                                                uint32_t* __restrict__ tf_part,
                                                float* __restrict__ tot_part) {
  __shared__ float smem[2][TROWS * K_];   // 2 x 32KB double buffer
  const int tid  = threadIdx.x;
  const int s    = blockIdx.x;
  const int b    = blockIdx.y;
  const int lane = tid & 31;
  const int wid  = tid >> 5;
  const int l_start = s * (L_ / S_);

  const uint32_t lds0 = (uint32_t)(uintptr_t)(&smem[0][0]) + (uint32_t)tid * 16u;
  const uint32_t lds1 = lds0 + (uint32_t)TILE_BYTES;
  const uint64_t g =
      (uint64_t)(uintptr_t)(spikes + ((size_t)b * L_ + (size_t)l_start) * K_) +
      (uint64_t)tid * 16u;

  async_tile_load(lds0, g);   // prologue: tile 0 -> buf0

  uint32_t tfirst = (uint32_t)L_;   // sentinel: never spiked -> L (matches reference)
  v8f acc0 = {};                    // WMMA accumulators: 2 x 16-column tiles per wave
  v8f acc1 = {};
  v2f ones; ones.x = 1.0f; ones.y = 1.0f;   // all-ones A => C rows = column sums of B
  const int cb = wid * 32;
  const int lo = lane & 15;
  const int hi = lane >> 4;

  for (int t = 0; t < TILES; ++t) {
    if (t + 1 < TILES) {
      async_tile_load(((t + 1) & 1) ? lds1 : lds0, g + (uint64_t)(t + 1) * TILE_BYTES);
      asm volatile("s_wait_asynccnt 0x4" ::: "memory");  // tile t done, t+1 in flight
    } else {
      asm volatile("s_wait_asynccnt 0x0" ::: "memory");
    }
    __syncthreads();
    const float* cur = smem[t & 1];
    const int lg = l_start + t * TROWS;

    // first-spike scan: thread tid owns column k = tid
    #pragma unroll
    for (int l = 0; l < TROWS; ++l) {
      float v = cur[l * K_ + tid];
      uint32_t cand = (v > THRv) ? (uint32_t)(lg + l) : (uint32_t)L_;
      tfirst = min(tfirst, cand);
    }

    // WMMA column-sum accumulation: ones(16x4) x spikes(4x16) tiles
    #pragma unroll
    for (int ks = 0; ks < 4; ++ks) {
      const int r = ks * 4 + hi * 2;
      v2f b0, b1;
      b0.x = cur[r * K_ + cb + lo];
      b0.y = cur[(r + 1) * K_ + cb + lo];
      b1.x = cur[r * K_ + cb + 16 + lo];
      b1.y = cur[(r + 1) * K_ + cb + 16 + lo];
      acc0 = __builtin_amdgcn_wmma_f32_16x16x4_f32(false, ones, false, b0,
                                                   (short)0, acc0, false, false);
      acc1 = __builtin_amdgcn_wmma_f32_16x16x4_f32(false, ones, false, b1,
                                                   (short)0, acc1, false, false);
    }
    __syncthreads();   // everyone done with this buffer before it is re-filled
  }

  const size_t base = ((size_t)s * B_ + (size_t)b) * K_;
  tf_part[base + tid] = tfirst;
  if (lane < 16) {   // C layout: VGPR0 lanes 0-15 hold N=lane (all M rows equal)
    tot_part[base + cb + lo]      = acc0[0];
    tot_part[base + cb + 16 + lo] = acc1[0];
  }
}

// ---------------- Pass 2: per-batch winner + straight-through weights ------
__global__ __launch_bounds__(512) void fs_pass2(const uint32_t* __restrict__ tf_part,
                                                const float* __restrict__ tot_part,
                                                float* __restrict__ out_idx,  // d_out[0..64)
                                                float* __restrict__ out_w) {  // d_out+64
  __shared__ uint32_t           sMin[512];
  __shared__ unsigned long long sMax[512];
  __shared__ float              sSum[512];
  const int k = threadIdx.x;
  const int b = blockIdx.x;

  uint32_t tf  = (uint32_t)L_;
  float    tot = 0.0f;
  for (int s = 0; s < S_; ++s) {
    const size_t base = ((size_t)s * B_ + (size_t)b) * K_ + (size_t)k;
    tf = min(tf, tf_part[base]);
    tot += tot_part[base];
  }

  // min t_first, tie -> lowest k (key = t*512 + k)
  sMin[k] = tf * 512u + (uint32_t)k;
  __syncthreads();
  for (int off = 256; off > 0; off >>= 1) {
    if (k < off) sMin[k] = min(sMin[k], sMin[k + off]);
    __syncthreads();
  }
  const uint32_t tmin  = sMin[0] >> 9;
  const int      kstar = (int)(sMin[0] & 511u);

  // fallback argmax(total), tie -> lowest k (tot >= 0 so float bits are monotone)
  sMax[k] = (((unsigned long long)__float_as_uint(tot)) << 9) |
            (unsigned long long)(511 - k);
  __syncthreads();
  for (int off = 256; off > 0; off >>= 1) {
    if (k < off) sMax[k] = max(sMax[k], sMax[k + off]);
    __syncthreads();
  }
  const int kfb = 511 - (int)(sMax[0] & 511ull);

  // softmax(-t_first/0.2) with max-shift (jax semantics)
  const float e = expf(-5.0f * (float)(tf - tmin));
  sSum[k] = e;
  __syncthreads();
  for (int off = 256; off > 0; off >>= 1) {
    if (k < off) sSum[k] += sSum[k + off];
    __syncthreads();
  }
  const float denom   = sSum[0];
  const bool  has_any = (tmin < (uint32_t)L_);
  const int   idx     = has_any ? kstar : kfb;

  const float wsur = e / denom;
  const float hard = (k == idx) ? 1.0f : 0.0f;
  float tmp = hard - wsur;
  asm volatile("" : "+v"(tmp));          // keep (hard - wsur) + wsur un-folded
  const float w = tmp + wsur;

  out_w[(size_t)b * K_ + k] = w;
  if (k == 0) out_idx[b] = (float)idx;
}

// ---------------- Pass 3: out3 = spikes * w  (exact zeros off-winner) ------
__global__ __launch_bounds__(256) void fs_pass3(const float* __restrict__ spikes,
                                                const float* __restrict__ out_head,
                                                float* __restrict__ out3) {
  __shared__ int   sidx;
  __shared__ float swv;
  const int tid  = threadIdx.x;
  const int b    = blockIdx.x >> 10;   // 1024 blocks per batch (uniform per block)
  if (tid == 0) {
    const int ii = (int)out_head[b];
    sidx = ii;
    swv  = out_head[64 + (size_t)b * K_ + ii];
  }
  __syncthreads();

  const size_t gid = (size_t)blockIdx.x * 256 + (size_t)tid;  // one float4 each
  const int    j   = (int)(gid & 127);
  const size_t row = gid >> 7;          // = b*2048 + l
  const int    ii  = sidx;

  v4f o = {0.0f, 0.0f, 0.0f, 0.0f};
  if (j == (ii >> 2)) o[ii & 3] = spikes[row * K_ + ii] * swv;
  __builtin_nontemporal_store(o, ((v4f*)out3) + gid);   // NT: don't pollute L2
}

// ---------------- Launch ---------------------------------------------------
extern "C" void kernel_launch(void* const* d_in, const int* in_sizes, int n_in,
                              void* d_out, int out_size, void* d_ws, size_t ws_size,
                              hipStream_t stream) {
  (void)in_sizes; (void)n_in; (void)d_ws; (void)ws_size;
  const float* spikes = (const float*)d_in[0];
  float* out = (float*)d_out;

  // Scratch lives in the tail of the out3 region of d_out: written by pass1,
  // read by pass2, then fully overwritten by pass3. No d_ws dependence.
  const size_t part_elems = (size_t)S_ * B_ * K_;          // 524288 each
  uint32_t* tf_part  = (uint32_t*)out + ((size_t)out_size - 2 * part_elems);
  float*    tot_part = out + ((size_t)out_size - part_elems);

  fs_pass1<<<dim3(S_, B_), 512, 0, stream>>>(spikes, tf_part, tot_part);
  fs_pass2<<<dim3(B_), 512, 0, stream>>>(tf_part, tot_part, out, out + 64);
  fs_pass3<<<dim3((B_ * (size_t)L_ * K_) / (4 * 256)), 256, 0, stream>>>(
      spikes, out, out + 64 + (size_t)B_ * K_);
}